// GCN2LayerMeanPool_26560077758926
// MI455X (gfx1250) — compile-verified
//
#include <hip/hip_runtime.h>

#define N_NODES 50000
#define N_EDGES 800000
#define FEAT 64
#define N_GRAPHS 128
#define OUTF 8

// padded LDS row stride for W: 80 floats => 2-row offset = 160 ≡ 32 (mod 64 banks),
// so lanes 0-15 (row k) and lanes 16-31 (row k+2) hit disjoint bank groups.
#define WSTRIDE 80

typedef __attribute__((ext_vector_type(2))) float v2f;
typedef __attribute__((ext_vector_type(8))) float v8f;

// ---------------------------------------------------------------- utilities
__global__ void zero_f32(float* __restrict__ p, long n) {
  long i = (long)blockIdx.x * blockDim.x + threadIdx.x;
  long stride = (long)gridDim.x * blockDim.x;
  for (; i < n; i += stride) p[i] = 0.0f;
}

__global__ void count_deg(const int* __restrict__ dst, float* __restrict__ deg, int E) {
  int i = blockIdx.x * blockDim.x + threadIdx.x;
  if (i < E) atomicAdd(&deg[dst[i]], 1.0f);
}

// dsq[i] = rsqrt(deg_i + 1)   (deg includes self loop)
__global__ void finalize_deg(const float* __restrict__ deg, float* __restrict__ dsq, int n) {
  int i = blockIdx.x * blockDim.x + threadIdx.x;
  if (i < n) dsq[i] = rsqrtf(deg[i] + 1.0f);
}

// ---------------------------------------------------------------- WMMA GEMM
// H[N x 64] = X[N x 64] @ W[64 x 64]  via V_WMMA_F32_16X16X4_F32 (full fp32).
// One wave32 per 16-row tile; 4 accumulators (16x16) cover all 64 columns.
// W is staged in LDS once per block (8 waves reuse it).
__global__ __launch_bounds__(256) void gemm64_wmma(const float* __restrict__ X,
                                                   const float* __restrict__ W,
                                                   float* __restrict__ H, int nrows) {
  __shared__ float sW[FEAT * WSTRIDE];  // 20 KB of the 320 KB WGP LDS

  // cooperative stage of W: 1024 float4 chunks across 256 threads (b128 ld/st)
  {
    const float4* Wv = (const float4*)W;
    float4* sWv = (float4*)sW;  // WSTRIDE % 4 == 0 -> row stride 20 float4
    for (int i = threadIdx.x; i < (FEAT * FEAT) / 4; i += 256) {
      const int r = i >> 4;        // row   (16 float4 per source row)
      const int c = i & 15;        // chunk within row
      sWv[r * (WSTRIDE / 4) + c] = Wv[i];
    }
  }
  __syncthreads();                 // all waves reach this before any exit

  const int wave = threadIdx.x >> 5;
  const int lane = threadIdx.x & 31;
  const int tile = blockIdx.x * 8 + wave;
  const int rowbase = tile * 16;
  if (rowbase >= nrows) return;    // wave-uniform: EXEC stays full below

  int mrow = rowbase + (lane & 15);
  if (mrow >= nrows) mrow = nrows - 1;  // clamp, keep all 32 lanes active
  const int hi = lane >> 4;             // 0: lanes 0-15, 1: lanes 16-31
  const int ncol = lane & 15;

  v8f acc[4] = {};

  // A 16x4 f32: VGPR0 holds K = k0 + 2*hi, VGPR1 holds K = k0 + 2*hi + 1.
  // B 4x16 f32: mirrored layout, rows from LDS.
  for (int k0 = 0; k0 < FEAT; k0 += 4) {
    const int ka = k0 + 2 * hi;
    v2f a;
    a[0] = X[(long)mrow * FEAT + ka];
    a[1] = X[(long)mrow * FEAT + ka + 1];
#pragma unroll
    for (int nt = 0; nt < 4; ++nt) {
      v2f b;
      b[0] = sW[ka * WSTRIDE + nt * 16 + ncol];
      b[1] = sW[(ka + 1) * WSTRIDE + nt * 16 + ncol];
      acc[nt] = __builtin_amdgcn_wmma_f32_16x16x4_f32(
          false, a, false, b, (short)0, acc[nt], false, false);
    }
  }

  // C/D layout: VGPR v -> M = v (lanes 0-15), M = v+8 (lanes 16-31); N = lane%16
  if (rowbase + 16 <= nrows) {     // full tile: branch-free stores (hot path)
#pragma unroll
    for (int nt = 0; nt < 4; ++nt)
#pragma unroll
      for (int v = 0; v < 8; ++v)
        H[(long)(rowbase + v + hi * 8) * FEAT + nt * 16 + ncol] = acc[nt][v];
  } else {                         // ragged tail (never taken for N=50000)
#pragma unroll
    for (int nt = 0; nt < 4; ++nt)
#pragma unroll
      for (int v = 0; v < 8; ++v) {
        const int r = rowbase + v + hi * 8;
        if (r < nrows) H[(long)r * FEAT + nt * 16 + ncol] = acc[nt][v];
      }
  }
}

// ---------------------------------------------------------------- aggregation
// agg[dst] += h[src] * dsq[src]*dsq[dst]   (thread per edge-feature pair)
__global__ void scatter_edges(const float* __restrict__ h,
                              const int* __restrict__ src,
                              const int* __restrict__ dst,
                              const float* __restrict__ dsq,
                              float* __restrict__ agg, int E) {
  long gid = (long)blockIdx.x * blockDim.x + threadIdx.x;
  if (gid >= (long)E * FEAT) return;
  const int e = (int)(gid >> 6);
  const int f = (int)(gid & 63);
  const int s = src[e];
  const int d = dst[e];
  const float norm = dsq[s] * dsq[d];
  atomicAdd(&agg[(long)d * FEAT + f], h[(long)s * FEAT + f] * norm);
}

// act = tanh(agg + h * (1/deg) + b),  1/deg = dsq^2
__global__ void combine_act(const float* __restrict__ agg,
                            const float* __restrict__ h,
                            const float* __restrict__ dsq,
                            const float* __restrict__ bias,
                            float* __restrict__ act, int n) {
  long gid = (long)blockIdx.x * blockDim.x + threadIdx.x;
  if (gid >= (long)n * FEAT) return;
  const int i = (int)(gid >> 6);
  const int f = (int)(gid & 63);
  const float ds = dsq[i];
  act[gid] = tanhf(agg[gid] + h[gid] * (ds * ds) + bias[f]);
}

// ---------------------------------------------------------------- pooling + FC
__global__ void pool_accum(const float* __restrict__ act,
                           const int* __restrict__ batch,
                           float* __restrict__ pooled,
                           float* __restrict__ counts, int n) {
  long gid = (long)blockIdx.x * blockDim.x + threadIdx.x;
  if (gid >= (long)n * FEAT) return;
  const int i = (int)(gid >> 6);
  const int f = (int)(gid & 63);
  const int g = batch[i];
  atomicAdd(&pooled[(long)g * FEAT + f], act[gid]);
  if (f == 0) atomicAdd(&counts[g], 1.0f);
}

__global__ void final_fc(const float* __restrict__ pooled,
                         const float* __restrict__ counts,
                         const float* __restrict__ Wfc,
                         const float* __restrict__ bfc,
                         float* __restrict__ out) {
  int gid = blockIdx.x * blockDim.x + threadIdx.x;
  if (gid >= N_GRAPHS * OUTF) return;
  const int g = gid / OUTF;
  const int o = gid % OUTF;
  float c = counts[g];
  c = (c > 1.0f) ? c : 1.0f;
  const float inv = 1.0f / c;
  float s = 0.0f;
#pragma unroll 8
  for (int f = 0; f < FEAT; ++f)
    s += pooled[(long)g * FEAT + f] * inv * Wfc[f * OUTF + o];
  out[gid] = s + bfc[o];
}

// ---------------------------------------------------------------- launcher
extern "C" void kernel_launch(void* const* d_in, const int* in_sizes, int n_in,
                              void* d_out, int out_size, void* d_ws, size_t ws_size,
                              hipStream_t stream) {
  (void)in_sizes; (void)n_in; (void)out_size; (void)ws_size;
  const float* x     = (const float*)d_in[0];
  const int*   src   = (const int*)d_in[1];
  const int*   dst   = (const int*)d_in[2];
  const int*   batch = (const int*)d_in[3];
  const float* W1    = (const float*)d_in[4];
  const float* b1    = (const float*)d_in[5];
  const float* W2    = (const float*)d_in[6];
  const float* b2    = (const float*)d_in[7];
  const float* Wfc   = (const float*)d_in[8];
  const float* bfc   = (const float*)d_in[9];
  float* out = (float*)d_out;

  float* ws     = (float*)d_ws;
  float* deg    = ws;                               // N
  float* dsq    = deg + N_NODES;                    // N
  float* h      = dsq + N_NODES;                    // N*64
  float* agg    = h   + (long)N_NODES * FEAT;       // N*64
  float* act    = agg + (long)N_NODES * FEAT;       // N*64
  float* pooled = act + (long)N_NODES * FEAT;       // G*64
  float* counts = pooled + N_GRAPHS * FEAT;         // G

  const long nf = (long)N_NODES * FEAT;
  const long ef = (long)N_EDGES * FEAT;
  const int  tiles = (N_NODES + 15) / 16;
  const int  gemmBlocks = (tiles + 7) / 8;

  // degrees + symmetric norms
  zero_f32<<<1024, 256, 0, stream>>>(deg, N_NODES);
  count_deg<<<(N_EDGES + 255) / 256, 256, 0, stream>>>(dst, deg, N_EDGES);
  finalize_deg<<<(N_NODES + 255) / 256, 256, 0, stream>>>(deg, dsq, N_NODES);

  // layer 1
  gemm64_wmma<<<gemmBlocks, 256, 0, stream>>>(x, W1, h, N_NODES);
  zero_f32<<<2048, 256, 0, stream>>>(agg, nf);
  scatter_edges<<<(int)((ef + 255) / 256), 256, 0, stream>>>(h, src, dst, dsq, agg, N_EDGES);
  combine_act<<<(int)((nf + 255) / 256), 256, 0, stream>>>(agg, h, dsq, b1, act, N_NODES);

  // layer 2
  gemm64_wmma<<<gemmBlocks, 256, 0, stream>>>(act, W2, h, N_NODES);
  zero_f32<<<2048, 256, 0, stream>>>(agg, nf);
  scatter_edges<<<(int)((ef + 255) / 256), 256, 0, stream>>>(h, src, dst, dsq, agg, N_EDGES);
  combine_act<<<(int)((nf + 255) / 256), 256, 0, stream>>>(agg, h, dsq, b2, act, N_NODES);

  // global mean pool + FC head
  zero_f32<<<64, 256, 0, stream>>>(pooled, (long)N_GRAPHS * FEAT);
  zero_f32<<<1, 128, 0, stream>>>(counts, N_GRAPHS);
  pool_accum<<<(int)((nf + 255) / 256), 256, 0, stream>>>(act, batch, pooled, counts, N_NODES);
  final_fc<<<(N_GRAPHS * OUTF + 255) / 256, 256, 0, stream>>>(pooled, counts, Wfc, bfc, out);
}